// SeedCondExpansion_67053029425411
// MI455X (gfx1250) — compile-verified
//
#include <hip/hip_runtime.h>
#include <hip/hip_bf16.h>
#include <stdint.h>

// ---------------------------------------------------------------------------
// Problem constants (from reference)
#define BB 16
#define NN 2048
#define DD 512
#define HH 256

typedef __attribute__((ext_vector_type(2))) float v2f;
typedef __attribute__((ext_vector_type(4))) float v4f;
typedef __attribute__((ext_vector_type(8))) float v8f;

// ---------------------------------------------------------------------------
// Kernel A: q = X_seed[16,512] @ Wq[512,256];  w = q[16,256] @ Wk^T -> [16,512]
// Both GEMMs on V_WMMA_F32_16X16X4_F32 (fp32 matrix pipe, M=16 fits one tile).
// One block, 32 waves. Phase 1: 16 waves make q tiles (N=256). Phase 2: all 32
// waves make w tiles (N=512).
__global__ __launch_bounds__(1024) void proj_wmma_kernel(
    const float* __restrict__ x, const int* __restrict__ seed_idx,
    const float* __restrict__ Wq, const float* __restrict__ Wk,
    float* __restrict__ w_out)
{
    __shared__ float sx[BB * DD];   // gathered seed rows, 32 KB
    __shared__ float qs[BB * HH];   // q, 16 KB
    const int t = threadIdx.x;

    for (int i = t; i < BB * DD; i += 1024) {
        int b = i >> 9, k = i & (DD - 1);
        sx[i] = x[((size_t)b * NN + (size_t)seed_idx[b]) * DD + k];
    }
    __syncthreads();

    const int wv   = t >> 5;
    const int lane = t & 31;
    const int half = lane >> 4;     // 0: lanes 0-15, 1: lanes 16-31
    const int l    = lane & 15;

    if (wv < 16) {                  // ---- phase 1: q tiles, N0 = wv*16
        const int n0 = wv * 16;
        v8f acc = {};
        for (int k0 = 0; k0 < DD; k0 += 4) {
            const int ka = k0 + half * 2;
            v2f a, bf;
            a.x  = sx[l * DD + ka];           // A[m=l][k=ka]
            a.y  = sx[l * DD + ka + 1];
            bf.x = Wq[(size_t)ka * HH + n0 + l];       // B[k=ka][n]
            bf.y = Wq[(size_t)(ka + 1) * HH + n0 + l];
            acc = __builtin_amdgcn_wmma_f32_16x16x4_f32(
                false, a, false, bf, (short)0, acc, false, false);
        }
        #pragma unroll
        for (int r = 0; r < 8; ++r)           // D: VGPR r -> rows r / r+8
            qs[(half * 8 + r) * HH + n0 + l] = acc[r];
    }
    __syncthreads();

    {                               // ---- phase 2: w tiles, N0 = wv*16 over 512
        const int n0 = wv * 16;
        v8f acc = {};
        for (int k0 = 0; k0 < HH; k0 += 4) {
            const int ka = k0 + half * 2;
            v2f a, bf;
            a.x  = qs[l * HH + ka];                    // A[m=l][k=ka] = q[b=l][h]
            a.y  = qs[l * HH + ka + 1];
            bf.x = Wk[(size_t)(n0 + l) * HH + ka];     // B[k=h][n=d] = Wk[d][h]
            bf.y = Wk[(size_t)(n0 + l) * HH + ka + 1];
            acc = __builtin_amdgcn_wmma_f32_16x16x4_f32(
                false, a, false, bf, (short)0, acc, false, false);
        }
        #pragma unroll
        for (int r = 0; r < 8; ++r)
            w_out[(half * 8 + r) * DD + n0 + l] = acc[r];
    }
}

// ---------------------------------------------------------------------------
// Kernel B: the bandwidth-bound pass. One wave per row n:
//   deg, e2c from one streamed NT read of adj[b,n,:]; xw = x[b,n,:] . w[b].
// cluster_mask[b,:] and w[b,:] are brought into LDS with CDNA5 async-to-LDS
// loads (ASYNCcnt-tracked, no VGPR round trip). adj (256 MB, > 192 MB L2,
// zero reuse) and x rows (single use) are loaded with TH=NT so they don't
// evict the genuinely reused data (cluster_mask rows, w, Wq/Wk) from L2.
// Dynamic LDS only => smem starts at LDS byte offset 0, so async VDST
// addresses are plain byte offsets.
__global__ __launch_bounds__(256) void fused_scores_kernel(
    const float* __restrict__ x, const float* __restrict__ adj,
    const float* __restrict__ cmask, const float* __restrict__ w,
    float* __restrict__ s_out)
{
    extern __shared__ float smem[];      // [0,8192): cm   [8192,10240): wv
    float* cm = smem;
    float* wvec = smem + NN;

    const int t    = threadIdx.x;
    const int row0 = blockIdx.x * 8;     // 8 waves/block, rows share b
    const int b    = row0 >> 11;

    // ---- async LDS fill: cm (2048 f32) + wvec (512 f32), b128 per lane ----
    for (int i = t; i < NN / 4; i += 256) {
        unsigned ldsoff = (unsigned)(i * 16);
        const float* g = cmask + (size_t)b * NN + i * 4;
        asm volatile("global_load_async_to_lds_b128 %0, %1, off"
                     :: "v"(ldsoff), "v"(g) : "memory");
    }
    if (t < DD / 4) {
        unsigned ldsoff = (unsigned)(NN * 4 + t * 16);
        const float* g = w + (size_t)b * DD + t * 4;
        asm volatile("global_load_async_to_lds_b128 %0, %1, off"
                     :: "v"(ldsoff), "v"(g) : "memory");
    }
    asm volatile("s_wait_asynccnt 0x0" ::: "memory");
    __syncthreads();

    const int wave = t >> 5, lane = t & 31;
    const int n = (row0 + wave) & (NN - 1);

    const v4f* __restrict__ arow = (const v4f*)(adj + ((size_t)b * NN + n) * NN);
    const v4f* __restrict__ xrow = (const v4f*)(x + ((size_t)b * NN + n) * DD);
    __builtin_prefetch(xrow + lane, 0, 0);   // global_prefetch for the x row

    float deg = 0.f, e2c = 0.f;
    #pragma unroll 4
    for (int i = 0; i < 16; ++i) {           // 2048 floats / 32 lanes / 4
        v4f v = __builtin_nontemporal_load(arow + lane + i * 32);
        int base = (lane + i * 32) * 4;
        deg += (v.x + v.y) + (v.z + v.w);
        e2c += v.x * cm[base] + v.y * cm[base + 1] +
               v.z * cm[base + 2] + v.w * cm[base + 3];
    }
    float xw = 0.f;
    #pragma unroll
    for (int i = 0; i < 4; ++i) {            // 512 floats / 32 lanes / 4
        v4f v = __builtin_nontemporal_load(xrow + lane + i * 32);
        int base = (lane + i * 32) * 4;
        xw += v.x * wvec[base] + v.y * wvec[base + 1] +
              v.z * wvec[base + 2] + v.w * wvec[base + 3];
    }
    #pragma unroll
    for (int off = 16; off >= 1; off >>= 1) { // wave32 butterfly
        deg += __shfl_xor(deg, off);
        e2c += __shfl_xor(e2c, off);
        xw  += __shfl_xor(xw, off);
    }
    if (lane == 0)
        s_out[(size_t)b * NN + n] = xw * 0.0625f /* 1/sqrt(256) */
                                  + 0.2f * e2c / fmaxf(deg, 1.0f);
}

// ---------------------------------------------------------------------------
// threefry2x32, bit-exact with jax.random.key(42): keys (0, 42)
#define TF_ROUND(r) { x0 += x1; x1 = (x1 << (r)) | (x1 >> (32 - (r))); x1 ^= x0; }
__device__ inline void threefry2x32_42(unsigned& x0, unsigned& x1) {
    const unsigned k0 = 0u, k1 = 42u;
    const unsigned k2 = k0 ^ k1 ^ 0x1BD11BDAu;
    x0 += k0; x1 += k1;
    TF_ROUND(13) TF_ROUND(15) TF_ROUND(26) TF_ROUND(6)
    x0 += k1; x1 += k2 + 1u;
    TF_ROUND(17) TF_ROUND(29) TF_ROUND(16) TF_ROUND(24)
    x0 += k2; x1 += k0 + 2u;
    TF_ROUND(13) TF_ROUND(15) TF_ROUND(26) TF_ROUND(6)
    x0 += k0; x1 += k1 + 3u;
    TF_ROUND(17) TF_ROUND(29) TF_ROUND(16) TF_ROUND(24)
    x0 += k1; x1 += k2 + 4u;
}

// Kernel C: mask + Gumbel noise (JAX split-counter layout: pair (p, p+16384)),
// also zero-initializes y_st (forward value of straight-through == y_hard).
__global__ __launch_bounds__(256) void gumbel_kernel(
    const float* __restrict__ scores, const int* __restrict__ cand,
    float* __restrict__ noisy, float* __restrict__ y_out)
{
    const int p = blockIdx.x * 256 + threadIdx.x;   // 0 .. B*N/2-1
    const int halfn = (BB * NN) / 2;
    if (p >= halfn) return;
    unsigned x0 = (unsigned)p, x1 = (unsigned)(p + halfn);
    threefry2x32_42(x0, x1);
    const unsigned outs[2] = { x0, x1 };
    #pragma unroll
    for (int h = 0; h < 2; ++h) {
        const int idx = p + h * halfn;
        const float u = __uint_as_float((outs[h] >> 9) | 0x3f800000u) - 1.0f;
        const float g = -logf(-logf(u + 1e-10f) + 1e-10f);
        const float m = cand[idx] ? scores[idx] : -3.4028234663852886e38f;
        noisy[idx] = m + g;
        y_out[idx] = 0.0f;
    }
}

// ---------------------------------------------------------------------------
// Kernel D: per-batch iterative top-K (K <= 63) over LDS-resident noisy row,
// lowest-index tie-break (matches lax.top_k). Writes int64 indices + y_hard.
__global__ __launch_bounds__(256) void topk_kernel(
    const float* __restrict__ noisy, const int* __restrict__ pred_K,
    int K_max, long long* __restrict__ idx_out, float* __restrict__ y_out)
{
    __shared__ float vals[NN];
    __shared__ float rv[256];
    __shared__ int   ri[256];
    const int b = blockIdx.x, t = threadIdx.x;
    for (int i = t; i < NN; i += 256) vals[i] = noisy[(size_t)b * NN + i];
    __syncthreads();
    const int Kb = pred_K[b];
    for (int j = 0; j < K_max; ++j) {
        float best = -__builtin_inff();
        int   bi   = 0x7fffffff;
        for (int i = t; i < NN; i += 256)
            if (vals[i] > best) { best = vals[i]; bi = i; }   // strict > => lowest idx
        rv[t] = best; ri[t] = bi;
        __syncthreads();
        for (int s = 128; s > 0; s >>= 1) {
            if (t < s) {
                if (rv[t+s] > rv[t] || (rv[t+s] == rv[t] && ri[t+s] < ri[t])) {
                    rv[t] = rv[t+s]; ri[t] = ri[t+s];
                }
            }
            __syncthreads();
        }
        if (t == 0) {
            const int sel = ri[0];
            idx_out[(size_t)b * K_max + j] = (long long)sel;
            if (j < Kb) y_out[(size_t)b * NN + sel] = 1.0f;
            vals[sel] = -__builtin_inff();    // exclude
        }
        __syncthreads();
    }
}

// ---------------------------------------------------------------------------
extern "C" void kernel_launch(void* const* d_in, const int* in_sizes, int n_in,
                              void* d_out, int out_size, void* d_ws, size_t ws_size,
                              hipStream_t stream) {
    (void)in_sizes; (void)n_in; (void)ws_size;
    const float* x        = (const float*)d_in[0];
    const float* adj      = (const float*)d_in[1];
    const int*   seed_idx = (const int*)  d_in[2];
    const float* cmask    = (const float*)d_in[3];
    const int*   cand     = (const int*)  d_in[4];
    const int*   pred_K   = (const int*)  d_in[5];
    const float* Wq       = (const float*)d_in[6];
    const float* Wk       = (const float*)d_in[7];

    // out = [topk_idx int64 (B*K_max)] ++ [y_st f32 (B*N)] ++ [scores f32 (B*N)]
    int K_max = (out_size - 2 * BB * NN) / BB;
    if (K_max < 1) K_max = 1;
    if (K_max > NN) K_max = NN;

    float* w_ws     = (float*)d_ws;          // B*D floats
    float* noisy_ws = w_ws + BB * DD;        // B*N floats

    long long* idx_out = (long long*)d_out;
    float*     y_out   = (float*)d_out + 2 * (size_t)BB * K_max;
    float*     s_out   = y_out + (size_t)BB * NN;

    proj_wmma_kernel<<<1, 1024, 0, stream>>>(x, seed_idx, Wq, Wk, w_ws);
    fused_scores_kernel<<<(BB * NN) / 8, 256, (NN + DD) * sizeof(float), stream>>>(
        x, adj, cmask, w_ws, s_out);
    gumbel_kernel<<<(BB * NN / 2 + 255) / 256, 256, 0, stream>>>(s_out, cand, noisy_ws, y_out);
    topk_kernel<<<BB, 256, 0, stream>>>(noisy_ws, pred_K, K_max, idx_out, y_out);
}